// AgnosticResidualInteractionBlock_51900384805553
// MI455X (gfx1250) — compile-verified
//
#include <hip/hip_runtime.h>
#include <math.h>

typedef float v2f __attribute__((ext_vector_type(2)));
typedef float v8f __attribute__((ext_vector_type(8)));

#define NN 16384
#define CC 128
#define EE 262144
#define NSPECIES 10
#define PERM_CAP (NN + NSPECIES * 16)   // worst-case padded permutation length

__device__ __forceinline__ float silu_f(float x) {
    return x / (1.0f + __expf(-x));
}

// ---------------------------------------------------------------------------
// Species bucketing: count -> aligned prefix -> fill. Each species region is
// padded to a multiple of 16 so every 16-row WMMA tile is single-species.
// ---------------------------------------------------------------------------
__global__ void k_sc_count(const int* __restrict__ specie, int* __restrict__ counters) {
    const int n = blockIdx.x * 256 + threadIdx.x;
    if (n < NN) atomicAdd(&counters[specie[n]], 1);
}

__global__ void k_sc_offsets(const int* __restrict__ counters,
                             int* __restrict__ off16, int* __restrict__ cursor) {
    if (threadIdx.x == 0) {
        int o = 0;
        for (int s = 0; s < NSPECIES; ++s) {
            off16[s]  = o;
            cursor[s] = o;
            o += (counters[s] + 15) & ~15;
        }
        off16[NSPECIES] = o;
    }
}

__global__ void k_sc_fill(const int* __restrict__ specie,
                          int* __restrict__ cursor, int* __restrict__ perm) {
    const int n = blockIdx.x * 256 + threadIdx.x;
    if (n < NN) {
        const int pos = atomicAdd(&cursor[specie[n]], 1);
        perm[pos] = n;
    }
}

// ---------------------------------------------------------------------------
// Kernel 1: species self-connection as dense WMMA GEMMs over species-sorted
// row tiles. A rows gathered via perm (padding rows masked to 0), B is the
// species' C x C weight matrix, D rows scattered back via perm.
// sc_s uses W_sc0 (comp 0), sc_v_i uses W_sc1 (comps 1..3).
// ---------------------------------------------------------------------------
__global__ void k_sc_gemm(const float* __restrict__ node_feats,
                          const float* __restrict__ W_sc0,
                          const float* __restrict__ W_sc1,
                          const int* __restrict__ off16,
                          const int* __restrict__ perm,
                          float* __restrict__ out_sc) {
    const int wg    = blockIdx.x * 8 + (threadIdx.x >> 5);
    const int lane  = threadIdx.x & 31;
    const int ntile = wg & 7;
    const int comp  = (wg >> 3) & 3;
    const int mtile = wg >> 5;
    const int m_base = mtile * 16;
    if (m_base >= off16[NSPECIES]) return;      // wave-uniform early exit

    int sp = 0;
    while (sp < NSPECIES - 1 && m_base >= off16[sp + 1]) ++sp;
    const float* W = ((comp == 0) ? W_sc0 : W_sc1) + sp * 16384;

    const int r    = lane & 15;
    const int half = lane >> 4;
    const int kh   = half * 2;
    const int ncol = ntile * 16 + r;

    const int   node = perm[m_base + r];
    const float amask = (node >= 0) ? 1.0f : 0.0f;
    const float* Arow = node_feats + (node >= 0 ? node : 0) * 512 + comp;

    v8f acc = {};
    for (int k0 = 0; k0 < 128; k0 += 4) {
        const int ka = k0 + kh;
        v2f a, b;
        a.x = Arow[ka * 4] * amask;
        a.y = Arow[(ka + 1) * 4] * amask;
        b.x = W[ka * 128 + ncol];
        b.y = W[(ka + 1) * 128 + ncol];
        acc = __builtin_amdgcn_wmma_f32_16x16x4_f32(false, a, false, b,
                                                    (short)0, acc, false, false);
    }
    const float inv_c = 0.08838834764831845f; // 1/sqrt(128)
    #pragma unroll
    for (int rr = 0; rr < 8; ++rr) {
        const int nd = perm[m_base + rr + half * 8];
        if (nd >= 0)
            out_sc[(nd * 128 + ncol) * 4 + comp] = acc[rr] * inv_c;
    }
}

// ---------------------------------------------------------------------------
// Kernel 2: pre-mix GEMMs with f32 WMMA.
// s = s_in @ W_pre0 * inv_c ; v_i = v_in_i @ W_pre1 * inv_c  (4 GEMMs,
// M=16384, K=128, N=128). One 16x16 output tile per wave. Output interleaved
// as sv[n][d][comp] so the edge kernel gathers one contiguous float4.
// ---------------------------------------------------------------------------
__global__ void k_pre_gemm(const float* __restrict__ node_feats,
                           const float* __restrict__ W_pre0,
                           const float* __restrict__ W_pre1,
                           float* __restrict__ sv) {
    const int wg    = blockIdx.x * 8 + (threadIdx.x >> 5);
    const int lane  = threadIdx.x & 31;
    const int ntile = wg & 7;
    const int comp  = (wg >> 3) & 3;
    const int mtile = wg >> 5;
    const float* W  = (comp == 0) ? W_pre0 : W_pre1;

    const int r    = lane & 15;
    const int half = lane >> 4;
    const int kh   = half * 2;
    const int mrow = mtile * 16 + r;
    const int ncol = ntile * 16 + r;

    const float* Arow = node_feats + mrow * 512 + comp;
    v8f acc = {};
    for (int k0 = 0; k0 < 128; k0 += 4) {
        const int ka = k0 + kh;
        v2f a, b;
        a.x = Arow[ka * 4];
        a.y = Arow[(ka + 1) * 4];
        b.x = W[ka * 128 + ncol];
        b.y = W[(ka + 1) * 128 + ncol];
        acc = __builtin_amdgcn_wmma_f32_16x16x4_f32(false, a, false, b,
                                                    (short)0, acc, false, false);
    }
    const float inv_c = 0.08838834764831845f;
    #pragma unroll
    for (int rr = 0; rr < 8; ++rr) {
        const int m = mtile * 16 + rr + half * 8;
        sv[(m * 128 + ncol) * 4 + comp] = acc[rr] * inv_c;
    }
}

// ---------------------------------------------------------------------------
// Kernel 3: edge MLP (weights staged once per block in LDS, 19.5 KB).
// mix[e,0:4] = silu(silu(ef@W1/sqrt8)@W2*0.125)@W3*0.125
// ---------------------------------------------------------------------------
__global__ void k_edge_mlp(const float* __restrict__ edge_feats,
                           const float* __restrict__ W1,
                           const float* __restrict__ W2,
                           const float* __restrict__ W3,
                           float* __restrict__ mix_out) {
    __shared__ float lw[4864];
    for (int i = threadIdx.x; i < 4864; i += 256)
        lw[i] = (i < 512) ? W1[i] : (i < 4608 ? W2[i - 512] : W3[i - 4608]);
    __syncthreads();
    const float* lW1 = lw;
    const float* lW2 = lw + 512;
    const float* lW3 = lw + 4608;

    const int e = blockIdx.x * 256 + threadIdx.x;
    float ef[8];
    #pragma unroll
    for (int k = 0; k < 8; ++k) ef[k] = edge_feats[e * 8 + k];

    float h2[64];
    #pragma unroll
    for (int j = 0; j < 64; ++j) h2[j] = 0.0f;

    const float inv_sqrt8 = 0.35355339059327373f;
    for (int j = 0; j < 64; ++j) {
        float h = 0.0f;
        #pragma unroll
        for (int k = 0; k < 8; ++k) h += ef[k] * lW1[k * 64 + j];
        h = silu_f(h * inv_sqrt8);
        #pragma unroll
        for (int j2 = 0; j2 < 64; ++j2) h2[j2] += h * lW2[j * 64 + j2];
    }
    float m0 = 0.f, m1 = 0.f, m2 = 0.f, m3 = 0.f;
    #pragma unroll
    for (int j2 = 0; j2 < 64; ++j2) {
        const float h = silu_f(h2[j2] * 0.125f);
        m0 += h * lW3[j2 * 4 + 0];
        m1 += h * lW3[j2 * 4 + 1];
        m2 += h * lW3[j2 * 4 + 2];
        m3 += h * lW3[j2 * 4 + 3];
    }
    ((float4*)mix_out)[e] = make_float4(m0 * 0.125f, m1 * 0.125f,
                                        m2 * 0.125f, m3 * 0.125f);
}

// ---------------------------------------------------------------------------
// Kernel 4: fused message build + scatter. msg (E,C,8) = 1 GB is never
// materialized. 2 edges per 256-thread block; one thread per (edge, channel)
// gathers a float4 of sender features (L2-resident sv) and does 8 f32
// atomicAdds into agg[receiver] (L2-resident, 67 MB).
// ---------------------------------------------------------------------------
__global__ void k_edge_scatter(const float* __restrict__ sv,
                               const float* __restrict__ mix,
                               const float* __restrict__ edge_attrs,
                               const int* __restrict__ senders,
                               const int* __restrict__ receivers,
                               float* __restrict__ agg) {
    const int e = blockIdx.x * 2 + (threadIdx.x >> 7);
    const int c = threadIdx.x & 127;
    const int snd = senders[e];
    const int rcv = receivers[e];
    const float ax = edge_attrs[e * 4 + 1];
    const float ay = edge_attrs[e * 4 + 2];
    const float az = edge_attrs[e * 4 + 3];
    const float4 m = ((const float4*)mix)[e];
    const float4 f = ((const float4*)sv)[snd * 128 + c];
    const float s = f.x;
    const float INV_SQRT3 = 0.5773502691896258f;
    const float dot = (f.y * ax + f.z * ay + f.w * az) * INV_SQRT3;
    float* base = agg + (rcv * 128 + c) * 8;
    atomicAdd(base + 0, s * m.x);
    atomicAdd(base + 1, dot * m.y);
    atomicAdd(base + 2, f.y * m.z);
    atomicAdd(base + 3, f.z * m.z);
    atomicAdd(base + 4, f.w * m.z);
    const float sm3 = s * m.w;
    atomicAdd(base + 5, sm3 * ax);
    atomicAdd(base + 6, sm3 * ay);
    atomicAdd(base + 7, sm3 * az);
}

// ---------------------------------------------------------------------------
// Kernel 5: post-mix GEMMs with f32 WMMA (4 GEMMs, M=16384, K=256, N=128).
// A[n,k] indexes agg (N,C,8): k=2c+m -> c*8 + {m | 2+3m+(comp-1)}.
// W_post* (C,2,C) == row-major (256,128). Scale folds in /16: 1/256.
// ---------------------------------------------------------------------------
__global__ void k_post_gemm(const float* __restrict__ agg,
                            const float* __restrict__ W_post0,
                            const float* __restrict__ W_post1,
                            float* __restrict__ node_out) {
    const int wg    = blockIdx.x * 8 + (threadIdx.x >> 5);
    const int lane  = threadIdx.x & 31;
    const int ntile = wg & 7;
    const int comp  = (wg >> 3) & 3;
    const int mtile = wg >> 5;
    const float* W  = (comp == 0) ? W_post0 : W_post1;

    const int r    = lane & 15;
    const int half = lane >> 4;
    const int kh   = half * 2;
    const int mrow = mtile * 16 + r;
    const int ncol = ntile * 16 + r;

    int sub0, sub1;
    if (comp == 0) { sub0 = 0; sub1 = 1; }
    else           { sub0 = 2 + (comp - 1); sub1 = 5 + (comp - 1); }

    const float* Arow = agg + mrow * 1024;
    v8f acc = {};
    for (int k0 = 0; k0 < 256; k0 += 4) {
        const int ka = k0 + kh;             // even
        v2f a, b;
        a.x = Arow[(ka >> 1) * 8 + sub0];   // k even  -> m=0
        a.y = Arow[(ka >> 1) * 8 + sub1];   // k+1 odd -> m=1, same c
        b.x = W[ka * 128 + ncol];
        b.y = W[(ka + 1) * 128 + ncol];
        acc = __builtin_amdgcn_wmma_f32_16x16x4_f32(false, a, false, b,
                                                    (short)0, acc, false, false);
    }
    const float scale = 0.00390625f; // 1/(16*sqrt(2*C))
    #pragma unroll
    for (int rr = 0; rr < 8; ++rr) {
        const int m = mtile * 16 + rr + half * 8;
        node_out[(m * 128 + ncol) * 4 + comp] = acc[rr] * scale;
    }
}

// ---------------------------------------------------------------------------
extern "C" void kernel_launch(void* const* d_in, const int* in_sizes, int n_in,
                              void* d_out, int out_size, void* d_ws, size_t ws_size,
                              hipStream_t stream) {
    const int*   node_specie = (const int*)d_in[0];
    const float* node_feats  = (const float*)d_in[1];
    const float* edge_attrs  = (const float*)d_in[2];
    const float* edge_feats  = (const float*)d_in[3];
    const int*   senders     = (const int*)d_in[4];
    const int*   receivers   = (const int*)d_in[5];
    const float* W_sc0   = (const float*)d_in[6];
    const float* W_sc1   = (const float*)d_in[7];
    const float* W_pre0  = (const float*)d_in[8];
    const float* W_pre1  = (const float*)d_in[9];
    const float* W_mlp1  = (const float*)d_in[10];
    const float* W_mlp2  = (const float*)d_in[11];
    const float* W_mlp3  = (const float*)d_in[12];
    const float* W_post0 = (const float*)d_in[13];
    const float* W_post1 = (const float*)d_in[14];

    float* out      = (float*)d_out;
    float* node_out = out;                       // (N,C,4) -- tuple elem 0
    float* out_sc   = out + (size_t)NN * CC * 4; // (N,C,4) -- tuple elem 1

    float* sv  = (float*)d_ws;                   // (N,C,4)  33.5 MB
    float* agg = sv + (size_t)NN * CC * 4;       // (N,C,8)  67 MB
    float* mix = agg + (size_t)NN * CC * 8;      // (E,4)    4 MB
    int*   ib  = (int*)(mix + (size_t)EE * 4);   // int scratch
    int* counters = ib;                          // [16]
    int* cursor   = ib + 16;                     // [16]
    int* off16    = ib + 32;                     // [16]
    int* perm     = ib + 64;                     // [PERM_CAP]

    hipMemsetAsync(agg, 0, (size_t)NN * CC * 8 * sizeof(float), stream);
    hipMemsetAsync(ib, 0, 64 * sizeof(int), stream);
    hipMemsetAsync(perm, 0xFF, (size_t)PERM_CAP * sizeof(int), stream);

    // species bucketing
    k_sc_count  <<<NN / 256, 256, 0, stream>>>(node_specie, counters);
    k_sc_offsets<<<1, 32, 0, stream>>>(counters, off16, cursor);
    k_sc_fill   <<<NN / 256, 256, 0, stream>>>(node_specie, cursor, perm);

    // species self-connection GEMM (wave-tiles: (PERM_CAP/16) m-tiles x 8 n x 4 comp)
    const int sc_blocks = (PERM_CAP / 16) * 8 * 4 / 8;
    k_sc_gemm<<<sc_blocks, 256, 0, stream>>>(node_feats, W_sc0, W_sc1,
                                             off16, perm, out_sc);

    k_pre_gemm<<<(NN / 16) * 8 * 4 / 8, 256, 0, stream>>>(node_feats, W_pre0, W_pre1, sv);
    k_edge_mlp<<<EE / 256, 256, 0, stream>>>(edge_feats, W_mlp1, W_mlp2, W_mlp3, mix);
    k_edge_scatter<<<EE / 2, 256, 0, stream>>>(sv, mix, edge_attrs, senders, receivers, agg);
    k_post_gemm<<<(NN / 16) * 8 * 4 / 8, 256, 0, stream>>>(agg, W_post0, W_post1, node_out);
}